// MultiHeadAttention_77223511982577
// MI455X (gfx1250) — compile-verified
//
#include <hip/hip_runtime.h>
#include <hip/hip_bf16.h>

typedef float v2f __attribute__((ext_vector_type(2)));
typedef float v8f __attribute__((ext_vector_type(8)));
typedef int   i32x4 __attribute__((vector_size(16)));

#define WMMA_F32(a, b, c) \
    __builtin_amdgcn_wmma_f32_16x16x4_f32(false, (a), false, (b), (short)0, (c), false, false)

#if defined(__gfx1250__) && __has_builtin(__builtin_amdgcn_global_load_async_to_lds_b128) && \
    __has_builtin(__builtin_amdgcn_s_wait_asynccnt)
#define HAVE_ASYNC_LDS 1
#else
#define HAVE_ASYNC_LDS 0
#endif

#if HAVE_ASYNC_LDS
// builtin signature (from probe): (v4i32 AS1*, v4i32 AS3*, imm offset, imm cpol)
#define ASYNC_CP16(gsrc, ldst)                                                        \
    __builtin_amdgcn_global_load_async_to_lds_b128(                                   \
        (i32x4 __attribute__((address_space(1)))*)(void*)(gsrc),                      \
        (i32x4 __attribute__((address_space(3)))*)(void*)(ldst), 0, 0)
#define ASYNC_WAIT() __builtin_amdgcn_s_wait_asynccnt(0)
#endif

constexpr int   Bsz   = 4;
constexpr int   Cdim  = 512;
constexpr int   Tlen  = 1024;
constexpr int   NH    = 8;
constexpr int   HD    = 64;
constexpr float SCALE_INV = 0.125f;   // 1/sqrt(64)

// LDS row stride (floats): even parity => 8B-aligned k-pairs; 68 % 64 == 4 =>
// bank(lane) = 4*lane + k, both k-halves of the wave hit disjoint banks.
constexpr int LS = 68;
// float2 pair-tile stride: 2*80 % 64 == 32 => the two k-halves are 32 banks apart.
constexpr int PS2 = 80;

__device__ __forceinline__ float rmax16(float x) {
    x = fmaxf(x, __shfl_xor(x, 1, 16));
    x = fmaxf(x, __shfl_xor(x, 2, 16));
    x = fmaxf(x, __shfl_xor(x, 4, 16));
    x = fmaxf(x, __shfl_xor(x, 8, 16));
    return x;
}
__device__ __forceinline__ float rsum16(float x) {
    x += __shfl_xor(x, 1, 16);
    x += __shfl_xor(x, 2, 16);
    x += __shfl_xor(x, 4, 16);
    x += __shfl_xor(x, 8, 16);
    return x;
}

// ---------------------------------------------------------------------------
// Kernel 1: fused Q/K/V projection.  out[b,o,t] = W[o,:] . x[b,:,t] + bias[o],
// written directly in head layout [B,H,T,d].
// grid = (T/64, C/64, B*3), block = 128 (4 waves, each wave: 16 rows x 64 cols)
// ---------------------------------------------------------------------------
__global__ __launch_bounds__(128)
void qkv_proj_kernel(const float* __restrict__ x,
                     const float* __restrict__ Wq, const float* __restrict__ bq,
                     const float* __restrict__ Wk, const float* __restrict__ bk,
                     const float* __restrict__ Wv, const float* __restrict__ bvv,
                     float* __restrict__ Q, float* __restrict__ Kg, float* __restrict__ Vg)
{
    const int tn = blockIdx.x;           // t tile
    const int tm = blockIdx.y;           // o tile
    const int b  = blockIdx.z / 3;
    const int pr = blockIdx.z % 3;
    const float* Wm   = (pr == 0) ? Wq : (pr == 1) ? Wk : Wv;
    const float* bias = (pr == 0) ? bq : (pr == 1) ? bk : bvv;
    float*       Out  = (pr == 0) ? Q  : (pr == 1) ? Kg : Vg;

    __shared__ float As[64][LS];     // W tile  [o][c]
    __shared__ v2f   Bp[32][PS2];    // x tile pairs: Bp[c/2][t] = {x[c][t], x[c+1][t]}

    const int tid  = threadIdx.x;
    const int wave = tid >> 5;
    const int lane = tid & 31;
    const int n    = lane & 15;          // N index within 16-tile
    const int kh   = (lane >> 4) * 2;    // K sub-offset (0 or 2)
    const int khp  = (lane >> 4);        // pair index sub-offset
    const int mhf  = (lane >> 4) * 8;    // C/D row-half base

    const float* xb = x + (size_t)b * Cdim * Tlen;

    v8f acc[4];
    #pragma unroll
    for (int j = 0; j < 4; ++j)
        #pragma unroll
        for (int v = 0; v < 8; ++v) acc[j][v] = 0.f;

    for (int kb = 0; kb < Cdim; kb += 64) {
        if (kb + 64 < Cdim) {
            __builtin_prefetch(Wm + (size_t)(tm * 64 + (tid >> 1)) * Cdim + kb + 64 + (tid & 1) * 32, 0, 0);
            __builtin_prefetch(xb + (size_t)(kb + 64 + (tid >> 1)) * Tlen + tn * 64 + (tid & 1) * 32, 0, 0);
        }
#if HAVE_ASYNC_LDS
        for (int i = tid; i < 1024; i += 128) {
            const int r  = i >> 4;
            const int c4 = (i & 15) << 2;
            ASYNC_CP16(Wm + (size_t)(tm * 64 + r) * Cdim + kb + c4, &As[r][c4]);
        }
#else
        for (int i = tid; i < 1024; i += 128) {
            const int r  = i >> 4;
            const int c4 = (i & 15) << 2;
            const float4 av = *(const float4*)(Wm + (size_t)(tm * 64 + r) * Cdim + kb + c4);
            As[r][c4 + 0] = av.x; As[r][c4 + 1] = av.y; As[r][c4 + 2] = av.z; As[r][c4 + 3] = av.w;
        }
#endif
        // pair-interleaved x tile: two rows fused per float2
        for (int i = tid; i < 1024; i += 128) {
            const int c2 = i >> 5;              // 0..31 row pair
            const int t2 = (i & 31) << 1;       // 0..62
            const float2 g1 = *(const float2*)(xb + (size_t)(kb + 2 * c2    ) * Tlen + tn * 64 + t2);
            const float2 g2 = *(const float2*)(xb + (size_t)(kb + 2 * c2 + 1) * Tlen + tn * 64 + t2);
            float4 pk; pk.x = g1.x; pk.y = g2.x; pk.z = g1.y; pk.w = g2.y;
            *(float4*)&Bp[c2][t2] = pk;
        }
#if HAVE_ASYNC_LDS
        ASYNC_WAIT();
#endif
        __syncthreads();

        #pragma unroll
        for (int kk = 0; kk < 16; ++kk) {
            const v2f a = *(const v2f*)&As[wave * 16 + n][kk * 4 + kh];
            #pragma unroll
            for (int j = 0; j < 4; ++j) {
                const v2f bf = Bp[2 * kk + khp][j * 16 + n];
                acc[j] = WMMA_F32(a, bf, acc[j]);
            }
        }
        __syncthreads();
    }

    #pragma unroll
    for (int j = 0; j < 4; ++j) {
        const int t = tn * 64 + j * 16 + n;
        #pragma unroll
        for (int v = 0; v < 8; ++v) {
            const int   o   = tm * 64 + wave * 16 + mhf + v;
            const float val = acc[j][v] + bias[o];
            Out[(((size_t)b * NH + (o >> 6)) * Tlen + t) * HD + (o & 63)] = val;
        }
    }
}

// ---------------------------------------------------------------------------
// Kernel 2: flash attention with windowed relative-position terms.
// grid = (T/64, B*H), block = 128 (each wave owns 16 query rows).
// K and V tiles stay in natural [s][d] layout (async-copyable); for S = Q.K^T
// the contraction dim of the B operand is d, contiguous => b64 pair loads.
// ---------------------------------------------------------------------------
__global__ __launch_bounds__(128)
void attn_kernel(const float* __restrict__ Q, const float* __restrict__ Kg,
                 const float* __restrict__ Vg, const int* __restrict__ mask,
                 const float* __restrict__ erk, const float* __restrict__ erv,
                 float* __restrict__ Oh)
{
    const int tq = blockIdx.x;
    const int bh = blockIdx.y;
    const int b  = bh >> 3;              // H == 8

    __shared__ float Qs[64][LS];         // Q tile, resident for whole kernel
    __shared__ float KsP[64][LS];        // K tile [s][d]; reused as P [m][s] after S
    __shared__ float Vs[64][LS];         // V tile [s][d]
    __shared__ float Ek[9][64];
    __shared__ float Ev[9][64];
    __shared__ float QE[4][16][9];       // q . emb_rel_k
    __shared__ float PB[4][16][10];      // banded attention probs
    __shared__ float RM[4][16];
    __shared__ float RL[4][16];

    const int tid  = threadIdx.x;
    const int wave = tid >> 5;
    const int lane = tid & 31;
    const int n    = lane & 15;
    const int kh   = (lane >> 4) * 2;
    const int mhf  = (lane >> 4) * 8;
    const int t0   = tq * 64;

    const float* Qbh = Q  + (size_t)bh * Tlen * HD;
    const float* Kbh = Kg + (size_t)bh * Tlen * HD;
    const float* Vbh = Vg + (size_t)bh * Tlen * HD;
    const int*   mb  = mask + (size_t)b * Tlen;

#if HAVE_ASYNC_LDS
    for (int i = tid; i < 1024; i += 128) {
        const int r = i >> 4, c4 = (i & 15) << 2;
        ASYNC_CP16(Qbh + (size_t)(t0 + r) * HD + c4, &Qs[r][c4]);
    }
    for (int i = tid; i < 144; i += 128) {
        const int r = i >> 4, c4 = (i & 15) << 2;
        ASYNC_CP16(erk + r * HD + c4, &Ek[r][c4]);
        ASYNC_CP16(erv + r * HD + c4, &Ev[r][c4]);
    }
    ASYNC_WAIT();
#else
    for (int i = tid; i < 1024; i += 128) {
        const int r = i >> 4, c4 = (i & 15) << 2;
        const float4 qv = *(const float4*)(Qbh + (size_t)(t0 + r) * HD + c4);
        Qs[r][c4 + 0] = qv.x; Qs[r][c4 + 1] = qv.y; Qs[r][c4 + 2] = qv.z; Qs[r][c4 + 3] = qv.w;
    }
    for (int i = tid; i < 144; i += 128) {   // 9*64/4 float4
        const int r = i >> 4, c4 = (i & 15) << 2;
        const float4 a = *(const float4*)(erk + r * HD + c4);
        Ek[r][c4 + 0] = a.x; Ek[r][c4 + 1] = a.y; Ek[r][c4 + 2] = a.z; Ek[r][c4 + 3] = a.w;
        const float4 c = *(const float4*)(erv + r * HD + c4);
        Ev[r][c4 + 0] = c.x; Ev[r][c4 + 1] = c.y; Ev[r][c4 + 2] = c.z; Ev[r][c4 + 3] = c.w;
    }
#endif
    __syncthreads();

    // qe[r][w] = q_row(r) . emb_rel_k[w]   (per-wave, lane-parallel over 16*9 pairs)
    for (int p = lane; p < 144; p += 32) {
        const int r = p / 9, w = p % 9;
        float s = 0.f;
        #pragma unroll 16
        for (int c = 0; c < HD; ++c) s += Qs[wave * 16 + r][c] * Ek[w][c];
        QE[wave][r][w] = s;
    }

    // persistent Q fragments (A-layout for f32 16x16x4): aligned b64 loads
    v2f qf[16];
    #pragma unroll
    for (int kk = 0; kk < 16; ++kk)
        qf[kk] = *(const v2f*)&Qs[wave * 16 + n][kk * 4 + kh];

    float m_i[8], l_i[8];
    v8f   oacc[4];
    #pragma unroll
    for (int v = 0; v < 8; ++v) { m_i[v] = -3.0e38f; l_i[v] = 0.f; }
    #pragma unroll
    for (int j = 0; j < 4; ++j)
        #pragma unroll
        for (int v = 0; v < 8; ++v) oacc[j][v] = 0.f;

    for (int sb = 0; sb < Tlen; sb += 64) {
        if (sb + 64 < Tlen) {
            __builtin_prefetch(Kbh + (size_t)(sb + 64) * HD + tid * 32, 0, 0);
            __builtin_prefetch(Vbh + (size_t)(sb + 64) * HD + tid * 32, 0, 0);
        }
#if HAVE_ASYNC_LDS
        for (int i = tid; i < 1024; i += 128) {
            const int r = i >> 4, c4 = (i & 15) << 2;
            ASYNC_CP16(Kbh + (size_t)(sb + r) * HD + c4, &KsP[r][c4]);
            ASYNC_CP16(Vbh + (size_t)(sb + r) * HD + c4, &Vs[r][c4]);
        }
        ASYNC_WAIT();
#else
        for (int i = tid; i < 1024; i += 128) {
            const int r = i >> 4, c4 = (i & 15) << 2;
            const float4 kv = *(const float4*)(Kbh + (size_t)(sb + r) * HD + c4);
            KsP[r][c4 + 0] = kv.x; KsP[r][c4 + 1] = kv.y; KsP[r][c4 + 2] = kv.z; KsP[r][c4 + 3] = kv.w;
            const float4 vv = *(const float4*)(Vbh + (size_t)(sb + r) * HD + c4);
            Vs[r][c4 + 0] = vv.x; Vs[r][c4 + 1] = vv.y; Vs[r][c4 + 2] = vv.z; Vs[r][c4 + 3] = vv.w;
        }
#endif
        __syncthreads();

        // S = Q K^T  (16x64 per wave): B[k][n] = K[n][k] -> contiguous pair load
        v8f sacc[4];
        #pragma unroll
        for (int j = 0; j < 4; ++j)
            #pragma unroll
            for (int v = 0; v < 8; ++v) sacc[j][v] = 0.f;
        #pragma unroll
        for (int kk = 0; kk < 16; ++kk) {
            const v2f a = qf[kk];
            #pragma unroll
            for (int j = 0; j < 4; ++j) {
                const v2f bf = *(const v2f*)&KsP[j * 16 + n][kk * 4 + kh];
                sacc[j] = WMMA_F32(a, bf, sacc[j]);
            }
        }
        __syncthreads();   // all waves finished reading K tile (KsP reused below)

        // scale + windowed rel-k bias + mask
        float sv[4][8];
        #pragma unroll
        for (int j = 0; j < 4; ++j) {
            const int s  = sb + j * 16 + n;
            const int mk = mb[s];
            #pragma unroll
            for (int v = 0; v < 8; ++v) {
                const int r   = mhf + v;
                const int dlt = s - (t0 + wave * 16 + r);
                float val = sacc[j][v];
                if (dlt >= -4 && dlt <= 4) val += QE[wave][r][dlt + 4];
                val *= SCALE_INV;
                if (mk == 0) val = -10000.f;
                sv[j][v] = val;
            }
        }

        // online softmax update (row = 16 lanes of one half-wave)
        #pragma unroll
        for (int v = 0; v < 8; ++v) {
            float mx = fmaxf(fmaxf(sv[0][v], sv[1][v]), fmaxf(sv[2][v], sv[3][v]));
            mx = rmax16(mx);
            const float nm    = fmaxf(m_i[v], mx);
            const float alpha = __expf(m_i[v] - nm);
            m_i[v] = nm;
            float ss = 0.f;
            #pragma unroll
            for (int j = 0; j < 4; ++j) {
                const float p = __expf(sv[j][v] - nm);
                sv[j][v] = p;
                ss += p;
            }
            ss = rsum16(ss);
            l_i[v] = l_i[v] * alpha + ss;
            #pragma unroll
            for (int j = 0; j < 4; ++j) oacc[j][v] *= alpha;
        }

        // stage P in the (now free) K region, per-wave private rows [m][s]
        #pragma unroll
        for (int j = 0; j < 4; ++j)
            #pragma unroll
            for (int v = 0; v < 8; ++v)
                KsP[wave * 16 + mhf + v][j * 16 + n] = sv[j][v];

        // O += P x V
        #pragma unroll
        for (int kk = 0; kk < 16; ++kk) {
            const v2f a = *(const v2f*)&KsP[wave * 16 + n][kk * 4 + kh];
            #pragma unroll
            for (int j = 0; j < 4; ++j) {
                v2f bf;
                bf.x = Vs[kk * 4 + kh    ][j * 16 + n];
                bf.y = Vs[kk * 4 + kh + 1][j * 16 + n];
                oacc[j] = WMMA_F32(a, bf, oacc[j]);
            }
        }
        __syncthreads();   // protect KsP / Vs before next tile load
    }

    // publish row stats (lane n==0 of each half owns rows mhf..mhf+7)
    if (n == 0) {
        #pragma unroll
        for (int v = 0; v < 8; ++v) { RM[wave][mhf + v] = m_i[v]; RL[wave][mhf + v] = l_i[v]; }
    }

    // banded probabilities: pband[r][w] = exp(score(t, t+w-4) - m)/l, w-4 in [-4,4]
    for (int p = lane; p < 144; p += 32) {
        const int r = p / 9, w = p % 9;
        const int t = t0 + wave * 16 + r;
        const int s = t + w - 4;
        float pv = 0.f;
        if ((unsigned)s < (unsigned)Tlen) {
            const float* krow = Kbh + (size_t)s * HD;
            float dot = 0.f;
            #pragma unroll 16
            for (int c = 0; c < HD; ++c) dot += Qs[wave * 16 + r][c] * krow[c];
            float val = (dot + QE[wave][r][w]) * SCALE_INV;
            if (mb[s] == 0) val = -10000.f;
            pv = __expf(val - RM[wave][r]) / RL[wave][r];
        }
        PB[wave][r][w] = pv;
    }

    // normalize, add banded rel-v term, write head-layout output
    #pragma unroll
    for (int j = 0; j < 4; ++j) {
        const int dd = j * 16 + n;
        #pragma unroll
        for (int v = 0; v < 8; ++v) {
            const int r = mhf + v;
            float val = oacc[j][v] / RL[wave][r];
            #pragma unroll
            for (int w = 0; w < 9; ++w) val += PB[wave][r][w] * Ev[w][dd];
            Oh[((size_t)bh * Tlen + (t0 + wave * 16 + r)) * HD + dd] = val;
        }
    }
}

// ---------------------------------------------------------------------------
// Kernel 3: output projection.  out[b,o,t] = Wo[o,:] . Ohead[b,:,t] + bo[o]
// Ohead layout: [B,H,T,d] with c = h*64+dd.   grid = (T/64, C/64, B)
// The contraction dim (dd) is contiguous in Ohead => pair tile via b64 loads.
// ---------------------------------------------------------------------------
__global__ __launch_bounds__(128)
void out_proj_kernel(const float* __restrict__ Oh, const float* __restrict__ Wo,
                     const float* __restrict__ bo, float* __restrict__ out)
{
    const int tn = blockIdx.x, tm = blockIdx.y, b = blockIdx.z;

    __shared__ float As[64][LS];     // Wo tile [o][c]
    __shared__ v2f   Bp[32][PS2];    // Bp[dd/2][t] = {Oh[..][t][dd], Oh[..][t][dd+1]}

    const int tid  = threadIdx.x;
    const int wave = tid >> 5;
    const int lane = tid & 31;
    const int n    = lane & 15;
    const int kh   = (lane >> 4) * 2;
    const int khp  = (lane >> 4);
    const int mhf  = (lane >> 4) * 8;

    v8f acc[4];
    #pragma unroll
    for (int j = 0; j < 4; ++j)
        #pragma unroll
        for (int v = 0; v < 8; ++v) acc[j][v] = 0.f;

    for (int kb = 0; kb < Cdim; kb += 64) {
        const int h = kb >> 6;
        const float* obase = Oh + (((size_t)b * NH + h) * Tlen + tn * 64) * HD;
        if (kb + 64 < Cdim)
            __builtin_prefetch(Wo + (size_t)(tm * 64 + (tid >> 1)) * Cdim + kb + 64 + (tid & 1) * 32, 0, 0);
#if HAVE_ASYNC_LDS
        for (int i = tid; i < 1024; i += 128) {
            const int r = i >> 4, c4 = (i & 15) << 2;
            ASYNC_CP16(Wo + (size_t)(tm * 64 + r) * Cdim + kb + c4, &As[r][c4]);
        }
#else
        for (int i = tid; i < 1024; i += 128) {
            const int r = i >> 4, c4 = (i & 15) << 2;
            const float4 av = *(const float4*)(Wo + (size_t)(tm * 64 + r) * Cdim + kb + c4);
            As[r][c4 + 0] = av.x; As[r][c4 + 1] = av.y; As[r][c4 + 2] = av.z; As[r][c4 + 3] = av.w;
        }
#endif
        for (int i = tid; i < 2048; i += 128) {
            const int dd2 = i >> 6;             // 0..31
            const int t   = i & 63;             // 0..63
            Bp[dd2][t] = *(const v2f*)(obase + (size_t)t * HD + 2 * dd2);
        }
#if HAVE_ASYNC_LDS
        ASYNC_WAIT();
#endif
        __syncthreads();

        #pragma unroll
        for (int kk = 0; kk < 16; ++kk) {
            const v2f a = *(const v2f*)&As[wave * 16 + n][kk * 4 + kh];
            #pragma unroll
            for (int j = 0; j < 4; ++j) {
                const v2f bf = Bp[2 * kk + khp][j * 16 + n];
                acc[j] = WMMA_F32(a, bf, acc[j]);
            }
        }
        __syncthreads();
    }

    #pragma unroll
    for (int j = 0; j < 4; ++j) {
        const int t = tn * 64 + j * 16 + n;
        #pragma unroll
        for (int v = 0; v < 8; ++v) {
            const int o = tm * 64 + wave * 16 + mhf + v;
            out[((size_t)b * Cdim + o) * Tlen + t] = acc[j][v] + bo[o];
        }
    }
}

// ---------------------------------------------------------------------------
extern "C" void kernel_launch(void* const* d_in, const int* in_sizes, int n_in,
                              void* d_out, int out_size, void* d_ws, size_t ws_size,
                              hipStream_t stream)
{
    const float* x   = (const float*)d_in[0];
    const int*   msk = (const int*)  d_in[1];
    const float* Wq  = (const float*)d_in[2];
    const float* bq  = (const float*)d_in[3];
    const float* Wk  = (const float*)d_in[4];
    const float* bk  = (const float*)d_in[5];
    const float* Wv  = (const float*)d_in[6];
    const float* bv  = (const float*)d_in[7];
    const float* Wo  = (const float*)d_in[8];
    const float* bo  = (const float*)d_in[9];
    const float* erk = (const float*)d_in[10];
    const float* erv = (const float*)d_in[11];

    float* ws = (float*)d_ws;
    const size_t plane = (size_t)Bsz * NH * Tlen * HD;   // 2M floats per tensor
    float* Q  = ws;
    float* K  = ws + plane;
    float* V  = ws + 2 * plane;
    float* Oh = ws + 3 * plane;

    dim3 blk(128);
    dim3 g1(Tlen / 64, Cdim / 64, Bsz * 3);
    qkv_proj_kernel<<<g1, blk, 0, stream>>>(x, Wq, bq, Wk, bk, Wv, bv, Q, K, V);

    dim3 g2(Tlen / 64, Bsz * NH);
    attn_kernel<<<g2, blk, 0, stream>>>(Q, K, V, msk, erk, erv, Oh);

    dim3 g3(Tlen / 64, Cdim / 64, Bsz);
    out_proj_kernel<<<g3, blk, 0, stream>>>(Oh, Wo, bo, (float*)d_out);
}